// BaseLayer_84155589198300
// MI455X (gfx1250) — compile-verified
//
#include <hip/hip_runtime.h>
#include <stdint.h>

// Feature dimension fixed at 128 in the reference: one float4 per lane of a wave32.
#define DFEAT 128
#define DV4   (DFEAT / 4)
#define WAVES_PER_BLOCK 8
#define BATCH 8   // edges staged + gathered per batch (8-deep vmem pipeline)

// ---------------------------------------------------------------------------
// gfx1250 async-copy helpers: per-lane global->LDS copy tracked by ASYNCcnt.
// asm is portable across ROCm 7.2 / amdgpu-toolchain builtin-arity skews and
// doubles as a compiler memory barrier around the staged-LDS accesses.
// Async completions are in-order per wave (CDNA5 ISA), so
// "s_wait_asynccnt 1" with two batches in flight guarantees the older one.
// ---------------------------------------------------------------------------
__device__ __forceinline__ void async_load_dword_to_lds(const int* gptr, int* lds_generic) {
    uint32_t lds_off = (uint32_t)(uintptr_t)lds_generic;   // addr[31:0] == LDS offset
    uint64_t gaddr   = (uint64_t)(uintptr_t)gptr;
    asm volatile("global_load_async_to_lds_b32 %0, %1, off"
                 :: "v"(lds_off), "v"(gaddr) : "memory");
}
__device__ __forceinline__ void wait_asynccnt0() {
    asm volatile("s_wait_asynccnt 0x0" ::: "memory");
}
__device__ __forceinline__ void wait_asynccnt1() {
    asm volatile("s_wait_asynccnt 0x1" ::: "memory");
}

// ---------------------------------------------------------------------------
// Kernel 1: zero the out-degree scratch (atomics accumulate; must re-zero
// every launch for determinism under graph replay).
// ---------------------------------------------------------------------------
__global__ void k_zero_i32(int* __restrict__ p, int n) {
    int i = blockIdx.x * blockDim.x + threadIdx.x;
    if (i < n) p[i] = 0;
}

// ---------------------------------------------------------------------------
// Kernel 2: out_deg[s] = #edges with src_idx == s  (scatter count)
// ---------------------------------------------------------------------------
__global__ void k_outdeg(const int* __restrict__ src_idx,
                         int* __restrict__ outdeg, int e) {
    int i = blockIdx.x * blockDim.x + threadIdx.x;
    if (i < e) atomicAdd(&outdeg[src_idx[i]], 1);
}

// ---------------------------------------------------------------------------
// Kernel 3: out_norm[s] = clip(out_deg,1)^-0.5 — hoists ~1M per-edge
// sqrt+div TRANS ops down to 50K; per-edge weight becomes a 4B broadcast load.
// ---------------------------------------------------------------------------
__global__ void k_outnorm(const int* __restrict__ outdeg,
                          float* __restrict__ outnorm, int n) {
    int i = blockIdx.x * blockDim.x + threadIdx.x;
    if (i < n) outnorm[i] = 1.0f / sqrtf(fmaxf((float)outdeg[i], 1.0f));
}

// ---------------------------------------------------------------------------
// Kernel 4: ptr[n] = lower_bound(dst_idx, n), n in [0, n_dst]; ptr[n_dst]==e.
// Replaces jnp.searchsorted; in_deg[n] = ptr[n+1]-ptr[n].
// ---------------------------------------------------------------------------
__global__ void k_ptr(const int* __restrict__ dst_idx,
                      int* __restrict__ ptr, int n_dst, int e) {
    int n = blockIdx.x * blockDim.x + threadIdx.x;
    if (n > n_dst) return;
    int lo = 0, hi = e;
    while (lo < hi) {
        int mid = (lo + hi) >> 1;
        if (dst_idx[mid] < n) lo = mid + 1; else hi = mid;
    }
    ptr[n] = lo;
}

// ---------------------------------------------------------------------------
// Kernel 5: main aggregation. One wave32 per dst node; lane l owns feature
// dims [4l,4l+4) so one wave-wide float4 load = one full 512B row
// (global_load_b128). 3-stage pipeline per wave:
//   [async index stage (ASYNCcnt)] || [8 row gathers (LOADcnt)] || [FMAs]
// with double-buffered LDS index staging and the first stage pre-issued
// before the category scan so its latency is fully hidden.
// ---------------------------------------------------------------------------
__global__ __launch_bounds__(WAVES_PER_BLOCK * 32) void k_aggregate(
    const float* __restrict__ h_src,
    const float* __restrict__ unif,
    const int*   __restrict__ src_idx,
    const int*   __restrict__ category,
    const float* __restrict__ outnorm,
    const int*   __restrict__ ptr,
    float*       __restrict__ out,
    int n_dst, int e_total, int kk)
{
    __shared__ int stage[WAVES_PER_BLOCK][2][BATCH];

    const int wave = threadIdx.x >> 5;
    const int lane = threadIdx.x & 31;
    const int node = blockIdx.x * WAVES_PER_BLOCK + wave;
    if (node >= n_dst) return;

    const int start = ptr[node];
    const int end   = ptr[node + 1];
    const int deg   = end - start;

    // Pre-issue batch 0 of the full-path indices into buffer 0; its latency
    // hides under the category scan below. (Speculative: harmless 32B if the
    // sampled path is taken — that path uses buffer 1.)
    if (deg > 0) {
        const int c0 = min(BATCH, deg);
        if (lane < c0)
            async_load_dword_to_lds(src_idx + start + lane, &stage[wave][0][lane]);
    }

    // ---- escape hatch: any incoming category == -1 forces the full sum ----
    int negl = 0;
    for (int e = start + lane; e < end; e += 32) {
        negl += (category[src_idx[e]] == -1) ? 1 : 0;
    }
    const bool has_neg  = (__any(negl > 0) != 0);
    const bool use_full = (deg <= kk) || has_neg;

    const float4* __restrict__ rows = (const float4*)h_src;   // row stride DV4
    float a0 = 0.f, a1 = 0.f, a2 = 0.f, a3 = 0.f;

    if (deg > 0) {
        if (use_full) {
            // -------- full segment sum, double-buffered batches of 8 --------
            int base = start;
            int buf  = 0;
            while (base < end) {
                const int  next_base = base + BATCH;
                const bool have_next = (next_base < end);   // wave-uniform
                if (have_next) {
                    const int ncnt = min(BATCH, end - next_base);
                    if (lane < ncnt)
                        async_load_dword_to_lds(src_idx + next_base + lane,
                                                &stage[wave][buf ^ 1][lane]);
                    if (lane < BATCH && next_base + BATCH + lane < end)
                        __builtin_prefetch(src_idx + next_base + BATCH + lane, 0, 0);
                    wait_asynccnt1();   // in-order: current batch has landed
                } else {
                    wait_asynccnt0();
                }
                const int cnt = min(BATCH, end - base);
                #pragma unroll
                for (int j = 0; j < BATCH; ++j) {
                    if (j < cnt) {
                        const int   s = stage[wave][buf][j];  // ds broadcast
                        const float w = outnorm[s];           // 4B broadcast load
                        const float4 r = rows[(size_t)s * DV4 + lane];
                        a0 = fmaf(r.x, w, a0);
                        a1 = fmaf(r.y, w, a1);
                        a2 = fmaf(r.z, w, a2);
                        a3 = fmaf(r.w, w, a3);
                    }
                }
                base = next_base;
                buf ^= 1;
            }
        } else {
            // -------- K random resamples with replacement (buffer 1) --------
            if (lane < kk) {
                const float u = unif[(size_t)node * kk + lane];
                int off = (int)floorf(u * (float)deg);
                off = min(off, deg - 1);
                off = max(off, 0);
                int eid = start + off;
                eid = min(max(eid, 0), e_total - 1);
                async_load_dword_to_lds(src_idx + eid, &stage[wave][1][lane]);
            }
            wait_asynccnt0();
            #pragma unroll
            for (int j = 0; j < BATCH; ++j) {
                if (j < kk) {
                    const int   s = stage[wave][1][j];
                    const float w = outnorm[s];
                    const float4 r = rows[(size_t)s * DV4 + lane];
                    a0 = fmaf(r.x, w, a0);
                    a1 = fmaf(r.y, w, a1);
                    a2 = fmaf(r.z, w, a2);
                    a3 = fmaf(r.w, w, a3);
                }
            }
        }
    }

    // dst-side norm: clip(in_deg,1)^-0.5; deg==0 row stays zero (matches ref).
    const float innorm = 1.0f / sqrtf(fmaxf((float)deg, 1.0f));
    float4 o;
    o.x = a0 * innorm;
    o.y = a1 * innorm;
    o.z = a2 * innorm;
    o.w = a3 * innorm;
    ((float4*)out)[(size_t)node * DV4 + lane] = o;
}

// ---------------------------------------------------------------------------
// Launcher (graph-capture safe: kernel launches only, scratch in d_ws).
// Input order: h_src, h_dst, unif, src_idx, dst_idx, category
// ---------------------------------------------------------------------------
extern "C" void kernel_launch(void* const* d_in, const int* in_sizes, int n_in,
                              void* d_out, int out_size, void* d_ws, size_t ws_size,
                              hipStream_t stream) {
    const float* h_src    = (const float*)d_in[0];
    // d_in[1] (h_dst) contributes only its shape; values unused by the forward.
    const float* unif     = (const float*)d_in[2];
    const int*   src_idx  = (const int*)d_in[3];
    const int*   dst_idx  = (const int*)d_in[4];
    const int*   category = (const int*)d_in[5];
    float*       out      = (float*)d_out;

    const int n_src = in_sizes[0] / DFEAT;
    const int n_dst = in_sizes[1] / DFEAT;
    const int e     = in_sizes[3];
    const int kk    = in_sizes[2] / n_dst;   // K = 8

    // Workspace: [ out_deg : n_src i32 ][ ptr : n_dst+1 i32 ][ out_norm : n_src f32 ]
    int*   outdeg  = (int*)d_ws;
    int*   ptr     = outdeg + n_src;
    float* outnorm = (float*)(ptr + n_dst + 1);

    const int T = 256;
    k_zero_i32<<<(n_src + T - 1) / T, T, 0, stream>>>(outdeg, n_src);
    k_outdeg  <<<(e + T - 1) / T, T, 0, stream>>>(src_idx, outdeg, e);
    k_outnorm <<<(n_src + T - 1) / T, T, 0, stream>>>(outdeg, outnorm, n_src);
    k_ptr     <<<(n_dst + 1 + T - 1) / T, T, 0, stream>>>(dst_idx, ptr, n_dst, e);

    const int blocks = (n_dst + WAVES_PER_BLOCK - 1) / WAVES_PER_BLOCK;
    k_aggregate<<<blocks, WAVES_PER_BLOCK * 32, 0, stream>>>(
        h_src, unif, src_idx, category, outnorm, ptr, out, n_dst, e, kk);
}